// AI4DEM_27599459844978
// MI455X (gfx1250) — compile-verified
//
#include <hip/hip_runtime.h>

// AI4DEM 5x5x5 contact-force stencil + Euler integration for MI455X (gfx1250).
// FP32-VALU-bound (per-pair rsq + masked select before the reduction), so WMMA
// is not applicable. CDNA5-specific path used: global_load_async_to_lds_b32
// (ASYNCcnt DMA) halo fill of a 48KB LDS tile + s_wait_asynccnt drain, and
// v_rsq_f32 for the per-pair normalization.

namespace {
constexpr int NXg = 128;
constexpr long long N3 = 128LL * 128LL * 128LL;
constexpr int TX = 8, TY = 8, TZ = 8;                    // cells per block
constexpr int PXd = TX + 4, PYd = TY + 4, PZd = TZ + 4;  // +2 halo each side
constexpr int PN = PXd * PYd * PZd;                      // 1728 padded cells
constexpr int NTHREADS = 256;                            // 8 waves (wave32)

// physics constants (match reference)
constexpr float CS   = 0.05f;
constexpr float KN   = 600000.0f;
constexpr float DT   = 0.001f;
constexpr float Gacc = 9.8f;
constexpr float MASS = 1.4137200f;        // 4/3*3.1416*CS^3*2700
constexpr float INV_MASS = 1.0f / MASS;
constexpr float ETA  = 280.62469f;        // 2*gamma*sqrt(KN*MASS/2), rest=0.5
}

// Async DMA one f32 from global memory into LDS (CDNA5 gfx1250 path).
// VDST VGPR carries the LDS byte address; 64-bit global address in VGPR pair.
// Tracked by ASYNCcnt; drained with s_wait_asynccnt.
__device__ __forceinline__ void async_copy_f32(unsigned lds_byte_addr, const float* gptr) {
    asm volatile("global_load_async_to_lds_b32 %0, %1, off"
                 :
                 : "v"(lds_byte_addr), "v"(gptr)
                 : "memory");
}

__global__ __launch_bounds__(NTHREADS)
void dem_force_integrate_kernel(const float* __restrict__ xg,
                                const float* __restrict__ yg,
                                const float* __restrict__ zg,
                                const float* __restrict__ vxg,
                                const float* __restrict__ vyg,
                                const float* __restrict__ vzg,
                                const float* __restrict__ mg,
                                float* __restrict__ out)
{
    // 7 fields x 12^3 padded tile = 48384 B of LDS (WGP has 320 KB)
    __shared__ float smem[7][PN];

    const int tid = threadIdx.x;
    const int gx0 = blockIdx.x * TX;
    const int gy0 = blockIdx.y * TY;
    const int gz0 = blockIdx.z * TZ;

    const float* src[7] = {xg, yg, zg, vxg, vyg, vzg, mg};

    // ---- halo fill: async DMA in-range cells, ds-store zeros for OOB ----
    for (int i = tid; i < PN; i += NTHREADS) {
        const int px = i % PXd;
        const int py = (i / PXd) % PYd;
        const int pz = i / (PXd * PYd);
        const int gx = gx0 + px - 2;
        const int gy = gy0 + py - 2;
        const int gz = gz0 + pz - 2;
        if (gx >= 0 && gx < NXg && gy >= 0 && gy < NXg && gz >= 0 && gz < NXg) {
            const long long gi = ((long long)gz * NXg + gy) * NXg + gx;
#pragma unroll
            for (int a = 0; a < 7; ++a) {
                // low 32 bits of a generic LDS pointer == LDS byte offset
                async_copy_f32((unsigned)(unsigned long long)&smem[a][i], src[a] + gi);
            }
        } else {
#pragma unroll
            for (int a = 0; a < 7; ++a) {
                smem[a][i] = 0.0f;   // zero padding (matches jnp.pad)
            }
        }
    }
    asm volatile("s_wait_asynccnt 0" ::: "memory");  // drain async DMA
    __syncthreads();                                  // + dscnt for zero fills

    // ---- per-cell pair loop over 124 neighbor offsets ----
    for (int c = tid; c < TX * TY * TZ; c += NTHREADS) {
        const int lx = c % TX;
        const int ly = (c / TX) % TY;
        const int lz = c / (TX * TY);
        const int pi = ((lz + 2) * PYd + (ly + 2)) * PXd + (lx + 2);

        const float xc  = smem[0][pi], yc  = smem[1][pi], zc  = smem[2][pi];
        const float vxc = smem[3][pi], vyc = smem[4][pi], vzc = smem[5][pi];
        const float mc  = smem[6][pi];
        const bool  mcpos = (mc > 0.0f);   // loop-invariant mask term

        float fxa = 0.0f, fya = 0.0f, fza = 0.0f;

#pragma unroll 1
        for (int dz = -2; dz <= 2; ++dz) {
#pragma unroll 1
            for (int dy = -2; dy <= 2; ++dy) {
#pragma unroll
                for (int dx = -2; dx <= 2; ++dx) {
                    if (dx == 0 && dy == 0 && dz == 0) continue;
                    const int ni = pi + (dz * PYd + dy) * PXd + dx;
                    const float xn  = smem[0][ni];
                    const float yn  = smem[1][ni];
                    const float zn  = smem[2][ni];
                    const float vxn = smem[3][ni];
                    const float vyn = smem[4][ni];
                    const float vzn = smem[5][ni];
                    const float mn  = smem[6][ni];

                    const float dxx = xc - xn - (float)dx * CS;
                    const float dyy = yc - yn - (float)dy * CS;
                    const float dzz = zc - zn - (float)dz * CS;
                    const float d2  = fmaf(dxx, dxx, fmaf(dyy, dyy, dzz * dzz));

                    // one v_rsq_f32 replaces sqrt + Newton-refined divide
                    const float inv  = __builtin_amdgcn_rsqf(fmaxf(d2, 1e-24f));
                    const float dist = d2 * inv;            // == sqrt(d2); 0 at d2==0
                    const float eps  = CS - dist;

                    // unnormalized relative-velocity dot, normalized once
                    const float vdot = fmaf(vxc - vxn, dxx,
                                       fmaf(vyc - vyn, dyy, (vzc - vzn) * dzz));
                    const float fmag = KN * eps - ETA * (vdot * inv);

                    const bool active =
                        (eps > 0.0f) & (mn > 0.0f) & mcpos & (d2 > 0.0f);
                    const float s = active ? fmag * inv : 0.0f;
                    fxa = fmaf(s, dxx, fxa);
                    fya = fmaf(s, dyy, fya);
                    fza = fmaf(s, dzz, fza);
                }
            }
        }

        // ---- semi-implicit Euler integration, gravity on z ----
        const long long gi =
            ((long long)(gz0 + lz) * NXg + (gy0 + ly)) * NXg + (gx0 + lx);
        const float vx2 = mc * (vxc + fxa * INV_MASS * DT);
        const float vy2 = mc * (vyc + fya * INV_MASS * DT);
        const float vz2 = mc * (vzc + (fza * INV_MASS - Gacc) * DT);
        out[0 * N3 + gi] = mc * (xc + vx2 * DT);
        out[1 * N3 + gi] = mc * (yc + vy2 * DT);
        out[2 * N3 + gi] = mc * (zc + vz2 * DT);
        out[3 * N3 + gi] = vx2;
        out[4 * N3 + gi] = vy2;
        out[5 * N3 + gi] = vz2;
    }
}

extern "C" void kernel_launch(void* const* d_in, const int* in_sizes, int n_in,
                              void* d_out, int out_size, void* d_ws, size_t ws_size,
                              hipStream_t stream) {
    (void)in_sizes; (void)n_in; (void)out_size; (void)d_ws; (void)ws_size;
    const float* xg  = (const float*)d_in[0];
    const float* yg  = (const float*)d_in[1];
    const float* zg  = (const float*)d_in[2];
    const float* vxg = (const float*)d_in[3];
    const float* vyg = (const float*)d_in[4];
    const float* vzg = (const float*)d_in[5];
    const float* mg  = (const float*)d_in[6];
    float* out = (float*)d_out;

    dim3 grid(NXg / TX, NXg / TY, NXg / TZ);   // 16x16x16 blocks
    dim3 block(NTHREADS);
    dem_force_integrate_kernel<<<grid, block, 0, stream>>>(
        xg, yg, zg, vxg, vyg, vzg, mg, out);
}